// EnergyCoulomb_2774548873945
// MI455X (gfx1250) — compile-verified
//
#include <hip/hip_runtime.h>

typedef float v2f __attribute__((ext_vector_type(2)));
typedef float v8f __attribute__((ext_vector_type(8)));

#define D_IN    1024
#define H_DIM   512
#define N_ATOMS 512
#define B_BATCH 32
#define ROWS    32     // atom rows per workgroup (2 WMMA M-tiles)
#define KC      512    // K-chunk staged in LDS (ROWS*KC*4 = 64 KB)

__device__ __forceinline__ float ssp(float x) {
    // shifted softplus: softplus(x) - ln2, numerically stable
    return fmaxf(x, 0.0f) + log1pf(expf(-fabsf(x))) - 0.69314718055994530942f;
}

// ---------------------------------------------------------------------------
// Prep: Wt[h][k] = combined transposed weights ([W1 | Wc1] columns as rows)
// so WMMA B-fragments become contiguous 8-byte loads per lane.
// ---------------------------------------------------------------------------
__global__ __launch_bounds__(256) void
transpose_weights(const float* __restrict__ W1, const float* __restrict__ Wc1,
                  float* __restrict__ Wt)
{
    const int idx = blockIdx.x * 256 + threadIdx.x;   // 1024*1024 elements
    const int h = idx >> 10, k = idx & 1023;
    Wt[idx] = (h < H_DIM) ? W1[k * H_DIM + h] : Wc1[k * H_DIM + (h - H_DIM)];
}

// ---------------------------------------------------------------------------
// Fused dual-head MLP with V_WMMA_F32_16X16X4_F32.
//   hidden[m,h] = rep[m,:] . Wt[h,:]   (fp32 WMMA, K=1024)
//   yi[m] = sum_h<512  ssp(hidden + b1[h])  * W2[h]
//   q[m]  = sum_h>=512 ssp(hidden + bc1[h]) * Wc2[h]
// A staged in LDS in fragment-swizzled order (conflict-free ds_load_b64).
// ---------------------------------------------------------------------------
__global__ __launch_bounds__(256) void
mlp_wmma_kernel(const float* __restrict__ rep,
                const float* __restrict__ Wt,
                const float* __restrict__ b1,  const float* __restrict__ W2,
                const float* __restrict__ bc1, const float* __restrict__ Wc2,
                float* __restrict__ yi_buf, float* __restrict__ q_buf)
{
    __shared__ float lds[ROWS * KC];   // 64 KB
    const int tid  = threadIdx.x;
    const int lane = tid & 31;
    const int wave = tid >> 5;
    const int half = lane >> 4;        // which 16-lane group
    const int r    = lane & 15;
    const int row0 = blockIdx.x * ROWS;

    float accY[2][8], accQ[2][8];
    #pragma unroll
    for (int mt = 0; mt < 2; ++mt)
        #pragma unroll
        for (int j = 0; j < 8; ++j) { accY[mt][j] = 0.f; accQ[mt][j] = 0.f; }

    for (int i = 0; i < 8; ++i) {
        const int nt = wave + 8 * i;          // h-tile 0..63 (wave-private)
        const int h0 = nt * 16;
        const int h  = h0 + r;
        const bool isQ = (h0 >= H_DIM);
        // B fragment base: lane needs Wt[h0+r][k0 + 2*half .. +1]
        const float* __restrict__ bRow = Wt + (size_t)h * D_IN + 2 * half;

        v8f acc0 = {0,0,0,0,0,0,0,0};
        v8f acc1 = {0,0,0,0,0,0,0,0};

        for (int kc = 0; kc < D_IN; kc += KC) {
            __syncthreads();
            // Stage A chunk swizzled into per-lane fragment order:
            // element (row,k): chunk c=k>>2, lane = (row&15)+16*((k>>1)&1), slot = k&1
            for (int idx = tid; idx < ROWS * KC; idx += 256) {
                const int rr = idx >> 9;             // KC == 512
                const int kk = idx & (KC - 1);       // consecutive tid -> coalesced k
                const float v = rep[(size_t)(row0 + rr) * D_IN + kc + kk];
                const int mt = rr >> 4, r2 = rr & 15;
                const int c = kk >> 2, hf = (kk >> 1) & 1, slot = kk & 1;
                lds[((mt * (KC / 4) + c) * 32 + (r2 + 16 * hf)) * 2 + slot] = v;
            }
            __syncthreads();

            const float* bp = bRow + kc;
            #pragma unroll 8
            for (int k0 = 0; k0 < KC; k0 += 4) {
                const v2f bfrag = *(const v2f*)(bp + k0);                 // global_load_b64
                const int ci = k0 >> 2;
                const v2f a0 = *(const v2f*)&lds[(ci * 32 + lane) * 2];            // ds_load_b64
                const v2f a1 = *(const v2f*)&lds[((KC / 4 + ci) * 32 + lane) * 2]; // ds_load_b64
                acc0 = __builtin_amdgcn_wmma_f32_16x16x4_f32(
                           false, a0, false, bfrag, (short)0, acc0, false, false);
                acc1 = __builtin_amdgcn_wmma_f32_16x16x4_f32(
                           false, a1, false, bfrag, (short)0, acc1, false, false);
            }
        }

        // Epilogue: C/D layout -> VGPR j, lanes<16: M=j, lanes>=16: M=j+8; N = lane&15
        const float bias = isQ ? bc1[h - H_DIM] : b1[h];
        const float w2v  = isQ ? Wc2[h - H_DIM] : W2[h];
        #pragma unroll
        for (int j = 0; j < 8; ++j) {
            const float v0 = ssp(acc0[j] + bias) * w2v;
            const float v1 = ssp(acc1[j] + bias) * w2v;
            if (isQ) { accQ[0][j] += v0; accQ[1][j] += v1; }
            else     { accY[0][j] += v0; accY[1][j] += v1; }
        }
    }

    // Reduce the 16 lanes (tile columns) of each half-wave independently.
    #pragma unroll
    for (int mt = 0; mt < 2; ++mt)
        #pragma unroll
        for (int j = 0; j < 8; ++j)
            #pragma unroll
            for (int m = 8; m >= 1; m >>= 1) {
                accY[mt][j] += __shfl_xor(accY[mt][j], m, 32);
                accQ[mt][j] += __shfl_xor(accQ[mt][j], m, 32);
            }

    __syncthreads();                 // A staging region no longer needed; reuse
    if (r == 0) {                    // lanes 0 (M=j) and 16 (M=j+8) of each wave
        #pragma unroll
        for (int mt = 0; mt < 2; ++mt)
            #pragma unroll
            for (int j = 0; j < 8; ++j) {
                const int a = mt * 16 + j + 8 * half;
                lds[(wave * 2 + 0) * 32 + a] = accY[mt][j];
                lds[(wave * 2 + 1) * 32 + a] = accQ[mt][j];
            }
    }
    __syncthreads();
    if (tid < 64) {                  // deterministic fixed-order cross-wave sum
        const int head = tid >> 5, a = tid & 31;
        float s = 0.f;
        #pragma unroll
        for (int w = 0; w < 8; ++w) s += lds[(w * 2 + head) * 32 + a];
        (head ? q_buf : yi_buf)[row0 + a] = s;
    }
}

// ---------------------------------------------------------------------------
// Per-batch Coulomb + masked atom sum.  One block per batch, all N=512 atoms
// resident in LDS.  ~0.13 GFLOP total: negligible vs the GEMM.
// ---------------------------------------------------------------------------
__global__ __launch_bounds__(256) void
coulomb_kernel(const float* __restrict__ R,
               const float* __restrict__ mask,
               const float* __restrict__ yi_buf,
               const float* __restrict__ q_buf,
               const float* __restrict__ b2,
               const float* __restrict__ bc2,
               float* __restrict__ out)
{
    __shared__ float sx[N_ATOMS], sy[N_ATOMS], sz[N_ATOMS], sq[N_ATOMS];
    __shared__ float red[256];
    const int b = blockIdx.x;
    const int tid = threadIdx.x;
    const float b2v = b2[0], bc2v = bc2[0];

    float ypart = 0.f;
    for (int n = tid; n < N_ATOMS; n += 256) {
        const size_t gi = (size_t)b * N_ATOMS + n;
        sx[n] = R[gi * 3 + 0];
        sy[n] = R[gi * 3 + 1];
        sz[n] = R[gi * 3 + 2];
        const float m = mask[gi];
        sq[n] = (m != 0.f) ? (q_buf[gi] + bc2v) : 0.f;  // masked atoms drop out of pairs
        ypart += (yi_buf[gi] + b2v) * m;
    }
    __syncthreads();

    float epart = 0.f;
    for (int i = tid; i < N_ATOMS; i += 256) {
        const float xi = sx[i], yi2 = sy[i], zi = sz[i], qi = sq[i];
        for (int j = 0; j < N_ATOMS; ++j) {
            if (j == i) continue;                       // keep excludes diagonal
            const float dx = xi - sx[j], dy = yi2 - sy[j], dz = zi - sz[j];
            const float d2 = dx * dx + dy * dy + dz * dz;
            const float t = 1e-5f + sqrtf(d2);
            epart += sq[j] * qi / (t * t);
        }
    }
    red[tid] = ypart + epart;
    __syncthreads();
    for (int s = 128; s > 0; s >>= 1) {
        if (tid < s) red[tid] += red[tid + s];
        __syncthreads();
    }
    if (tid == 0) out[b] = red[0];
}

// ---------------------------------------------------------------------------
extern "C" void kernel_launch(void* const* d_in, const int* in_sizes, int n_in,
                              void* d_out, int out_size, void* d_ws, size_t ws_size,
                              hipStream_t stream)
{
    const float* rep  = (const float*)d_in[0];
    const float* R    = (const float*)d_in[1];
    const float* mask = (const float*)d_in[2];
    const float* W1   = (const float*)d_in[3];
    const float* b1   = (const float*)d_in[4];
    const float* W2   = (const float*)d_in[5];
    const float* b2   = (const float*)d_in[6];
    const float* Wc1  = (const float*)d_in[7];
    const float* bc1  = (const float*)d_in[8];
    const float* Wc2  = (const float*)d_in[9];
    const float* bc2  = (const float*)d_in[10];
    float* out = (float*)d_out;

    // Workspace: Wt (4 MB) | yi (64 KB) | q (64 KB)  ~= 4.2 MB total
    float* Wt     = (float*)d_ws;
    float* yi_buf = Wt + 1024 * 1024;
    float* q_buf  = yi_buf + B_BATCH * N_ATOMS;

    transpose_weights<<<(1024 * 1024) / 256, 256, 0, stream>>>(W1, Wc1, Wt);
    mlp_wmma_kernel<<<(B_BATCH * N_ATOMS) / ROWS, 256, 0, stream>>>(
        rep, Wt, b1, W2, bc1, Wc2, yi_buf, q_buf);
    coulomb_kernel<<<B_BATCH, 256, 0, stream>>>(R, mask, yi_buf, q_buf, b2, bc2, out);
}